// AggregPolicy_6811818131653
// MI455X (gfx1250) — compile-verified
//
#include <hip/hip_runtime.h>
#include <hip/hip_bf16.h>

typedef float v2f __attribute__((ext_vector_type(2)));
typedef float v8f __attribute__((ext_vector_type(8)));

// ---------------------------------------------------------------------------
// tanh: prefer the gfx1250 hardware transcendental; fallback is branch-free
// exp-based (saturates correctly at +/-1 for |x| large).
// ---------------------------------------------------------------------------
__device__ __forceinline__ float fast_tanh(float x) {
#if __has_builtin(__builtin_amdgcn_tanhf)
  return __builtin_amdgcn_tanhf(x);
#else
  float e = __expf(2.0f * x);
  return 1.0f - 2.0f * __builtin_amdgcn_rcpf(e + 1.0f);
#endif
}

// lanes 0-15 keep a; lanes 16-31 receive b's lane-(L-16) value
__device__ __forceinline__ float merge_lo_up(float a, float b, bool hi) {
  float sw = __shfl_xor(b, 16, 32);
  return hi ? sw : a;
}
// lanes 0-15 receive a's lane-(L+16) value; lanes 16-31 keep b
__device__ __forceinline__ float merge_hi_dn(float a, float b, bool hi) {
  float sw = __shfl_xor(a, 16, 32);
  return hi ? b : sw;
}

// ---------------------------------------------------------------------------
// Setup: assemble the banded 32x32 recurrence matrix M and fused bias into ws.
// State blocks: block 0 = h_m, block 1+n = h_j[n].
//   out block 0:   in0 = Whh_m, in1 = Wih_m
//   out block 1+n: in n = Wih_j (prev), in n+1 = Whh_j (self),
//                  in n+2 = Wih_j (next, if n<6)
// Bias: block0 = bih_m+bhh_m, blocks 1..7 = bih_j+bhh_j.
// ---------------------------------------------------------------------------
__global__ void build_mat_kernel(
    const float* __restrict__ Wih_j, const float* __restrict__ Whh_j,
    const float* __restrict__ bih_j, const float* __restrict__ bhh_j,
    const float* __restrict__ Wih_m, const float* __restrict__ Whh_m,
    const float* __restrict__ bih_m, const float* __restrict__ bhh_m,
    float* __restrict__ Mmat, float* __restrict__ bias32)
{
  int t  = threadIdx.x;          // 0..1023 == (m,k)
  int m  = t >> 5, k = t & 31;
  int br = m >> 2, orr = m & 3;
  int bc = k >> 2, ic  = k & 3;
  float v = 0.0f;
  if (br == 0) {
    if (bc == 0)      v = Whh_m[orr * 4 + ic];
    else if (bc == 1) v = Wih_m[orr * 4 + ic];
  } else {
    int n = br - 1;
    if (bc == n)               v += Wih_j[orr * 4 + ic];
    if (bc == n + 1)           v += Whh_j[orr * 4 + ic];
    if (n < 6 && bc == n + 2)  v += Wih_j[orr * 4 + ic];
  }
  Mmat[m * 32 + k] = v;
  if (k == 0)
    bias32[m] = (br == 0) ? (bih_m[orr] + bhh_m[orr]) : (bih_j[orr] + bhh_j[orr]);
}

// ---------------------------------------------------------------------------
// Main kernel: one wave32 handles 16 batch rows.
//   B operand (state): chunk c, VGPR0/1 = k=4c,4c+1 (lanes 0-15, row=lane)
//                      and k=4c+2,4c+3 (lanes 16-31, row=lane-16).
//   A operand (weights): tile(mbase,c): lane L<16 -> M=mbase+L, {K=4c,4c+1};
//                      L>=16 -> M=mbase+L-16, {K=4c+2,4c+3}  => one float2 load.
//   D: VGPR r holds out m=r (lanes 0-15) / m=r+8 (lanes 16-31), row = lane%16.
// Banded matrix: D0 needs chunks 0..4, D1 needs chunks 3..7 -> 10 WMMA / iter.
// ---------------------------------------------------------------------------
__global__ __launch_bounds__(256) void aggreg_wmma_kernel(
    const float* __restrict__ x,
    const float* __restrict__ Wj,   const float* __restrict__ bj,
    const float* __restrict__ Wm,   const float* __restrict__ bm,
    const float* __restrict__ Wact, const float* __restrict__ bact,
    const float* __restrict__ Mmat, const float* __restrict__ bias32,
    float* __restrict__ out, int B)
{
  const int  lane = threadIdx.x & 31;
  const int  nidx = lane & 15;
  const bool hi   = lane >= 16;
  const int  h0   = hi ? 2 : 0;
  const int  wave = threadIdx.x >> 5;

  const long row  = (long)(blockIdx.x * (blockDim.x >> 5) + wave) * 16 + nidx;
  const long rowc = row < (long)B ? row : (long)(B - 1);   // clamp: EXEC must stay full for WMMA

  // ---- constant A tiles (banded) ----
  v2f A0[5], A1[5];
  {
    const float* Ma = Mmat + nidx * 32 + h0;         // rows 0..15  -> D0
    const float* Mb = Mmat + (16 + nidx) * 32 + h0;  // rows 16..31 -> D1
#pragma unroll
    for (int c = 0; c < 5; ++c) {
      A0[c] = *(const v2f*)(Ma + 4 * c);             // chunks 0..4
      A1[c] = *(const v2f*)(Mb + 4 * (c + 3));       // chunks 3..7
    }
  }

  // ---- bias in D layout: D vgpr v -> out m = v + 8*hi (+16 for D1) ----
  float bD0[8], bD1[8];
#pragma unroll
  for (int v = 0; v < 8; ++v) {
    bD0[v] = bias32[v + (hi ? 8 : 0)];
    bD1[v] = bias32[16 + v + (hi ? 8 : 0)];
  }

  // ---- load input row (both lane halves load the same row) ----
  float xr[18];
  {
    const float* xp = x + rowc * 18;
#pragma unroll
    for (int i = 0; i < 18; ++i) xr[i] = xp[i];
  }

  // ---- initial state directly in B layout; lane computes comps h0,h0+1 ----
  v2f Bv[8];
#pragma unroll
  for (int t = 0; t < 2; ++t) {           // block 0: h_m = Wm @ obs + bm
    int h = h0 + t;
    float v = bm[h];
#pragma unroll
    for (int i = 0; i < 4; ++i) v += Wm[h * 4 + i] * xr[i];
    Bv[0][t] = v;
  }
#pragma unroll
  for (int n = 0; n < 7; ++n) {           // blocks 1..7: h_j[n] = Wj @ [j,jd] + bj
#pragma unroll
    for (int t = 0; t < 2; ++t) {
      int h = h0 + t;
      Bv[1 + n][t] = bj[h] + Wj[h * 2 + 0] * xr[4 + n] + Wj[h * 2 + 1] * xr[11 + n];
    }
  }

  // ---- 7 recurrent steps: s = tanh(M s + b) ----
  for (int it = 0; it < 7; ++it) {
    v8f acc0 = {}, acc1 = {};
#pragma unroll
    for (int c = 0; c < 5; ++c)
      acc0 = __builtin_amdgcn_wmma_f32_16x16x4_f32(
          false, A0[c], false, Bv[c], (short)0, acc0, false, false);
#pragma unroll
    for (int c = 0; c < 5; ++c)
      acc1 = __builtin_amdgcn_wmma_f32_16x16x4_f32(
          false, A1[c], false, Bv[3 + c], (short)0, acc1, false, false);

    float t0[8], t1[8];
#pragma unroll
    for (int v = 0; v < 8; ++v) {
      t0[v] = fast_tanh(acc0[v] + bD0[v]);
      t1[v] = fast_tanh(acc1[v] + bD1[v]);
    }

    // D layout -> B layout: half-wave swaps only (batch stays in lanes).
    Bv[0][0] = merge_lo_up(t0[0], t0[2], hi);  Bv[0][1] = merge_lo_up(t0[1], t0[3], hi);
    Bv[1][0] = merge_lo_up(t0[4], t0[6], hi);  Bv[1][1] = merge_lo_up(t0[5], t0[7], hi);
    Bv[2][0] = merge_hi_dn(t0[0], t0[2], hi);  Bv[2][1] = merge_hi_dn(t0[1], t0[3], hi);
    Bv[3][0] = merge_hi_dn(t0[4], t0[6], hi);  Bv[3][1] = merge_hi_dn(t0[5], t0[7], hi);
    Bv[4][0] = merge_lo_up(t1[0], t1[2], hi);  Bv[4][1] = merge_lo_up(t1[1], t1[3], hi);
    Bv[5][0] = merge_lo_up(t1[4], t1[6], hi);  Bv[5][1] = merge_lo_up(t1[5], t1[7], hi);
    Bv[6][0] = merge_hi_dn(t1[0], t1[2], hi);  Bv[6][1] = merge_hi_dn(t1[1], t1[3], hi);
    Bv[7][0] = merge_hi_dn(t1[4], t1[6], hi);  Bv[7][1] = merge_hi_dn(t1[5], t1[7], hi);
  }

  // ---- output: acts[n] = Wact . h_j[n] + bact (cross-half reduction) ----
  const float wa0 = Wact[h0], wa1 = Wact[h0 + 1];
  const float bb  = bact[0];
#pragma unroll
  for (int n = 0; n < 7; ++n) {
    float p = wa0 * Bv[1 + n][0] + wa1 * Bv[1 + n][1];
    float q = __shfl_xor(p, 16, 32);
    float a = p + q + bb;
    if (!hi && row < (long)B) out[row * 7 + n] = a;
  }
}

// ---------------------------------------------------------------------------
extern "C" void kernel_launch(void* const* d_in, const int* in_sizes, int n_in,
                              void* d_out, int out_size, void* d_ws, size_t ws_size,
                              hipStream_t stream) {
  (void)n_in; (void)out_size; (void)ws_size;
  const float* x     = (const float*)d_in[0];
  const float* Wj    = (const float*)d_in[1];
  const float* bj    = (const float*)d_in[2];
  const float* Wm    = (const float*)d_in[3];
  const float* bm    = (const float*)d_in[4];
  const float* Wih_j = (const float*)d_in[5];
  const float* Whh_j = (const float*)d_in[6];
  const float* bih_j = (const float*)d_in[7];
  const float* bhh_j = (const float*)d_in[8];
  const float* Wih_m = (const float*)d_in[9];
  const float* Whh_m = (const float*)d_in[10];
  const float* bih_m = (const float*)d_in[11];
  const float* bhh_m = (const float*)d_in[12];
  const float* Wact  = (const float*)d_in[13];
  const float* bact  = (const float*)d_in[14];

  const int B = in_sizes[0] / 18;

  float* Mmat   = (float*)d_ws;          // 32*32 floats
  float* bias32 = Mmat + 32 * 32;        // 32 floats

  build_mat_kernel<<<1, 1024, 0, stream>>>(Wih_j, Whh_j, bih_j, bhh_j,
                                           Wih_m, Whh_m, bih_m, bhh_m,
                                           Mmat, bias32);

  const int  wavesPerBlock = 8;                       // 256 threads
  const long tiles  = ((long)B + 15) / 16;            // 16 rows per wave
  const int  blocks = (int)((tiles + wavesPerBlock - 1) / wavesPerBlock);

  aggreg_wmma_kernel<<<blocks, 256, 0, stream>>>(x, Wj, bj, Wm, bm, Wact, bact,
                                                 Mmat, bias32, (float*)d_out, B);
}